// Learnable4DPE_1649267442334
// MI455X (gfx1250) — compile-verified
//
#include <hip/hip_runtime.h>

typedef float v2f __attribute__((ext_vector_type(2)));
typedef float v4f __attribute__((ext_vector_type(4)));
typedef float v8f __attribute__((ext_vector_type(8)));

#define PB    32
#define PC    128
#define PT    64
#define PE    512
#define PNPOS 256

// ---------------------------------------------------------------------------
// Kernel 1: nearest-neighbor indices via V_WMMA_F32_16X16X4_F32.
// One wave (32 lanes) handles a tile of 16 query points against all 256
// candidate positions (16 chunks of 16). A = -2 * pos (16x4, K={x,y,z,0}),
// B = positions^T chunk (4x16). C tile gives -2*q.p; adding |p|^2 (per-column,
// held by the lane that loaded it) gives the argmin-equivalent score.
// ---------------------------------------------------------------------------
__global__ __launch_bounds__(32) void nearest_idx_wmma(
    const float* __restrict__ pos,        // (B*C, 3)
    const float* __restrict__ positions,  // (NPOS, 3)
    int* __restrict__ idx_out)            // (B*C)
{
  const int lane = threadIdx.x;
  const int m  = lane & 15;   // row within half (A: M; B/C: N)
  const int hi = lane >> 4;   // 0 = lanes 0-15, 1 = lanes 16-31
  const int q0 = blockIdx.x * 16;

  // A 16x4 fp32 layout: lanes 0-15 hold K=0 (v0), K=1 (v1); lanes 16-31 K=2,K=3
  const float px = pos[(q0 + m) * 3 + 0];
  const float py = pos[(q0 + m) * 3 + 1];
  const float pz = pos[(q0 + m) * 3 + 2];
  v2f a;
  if (hi == 0) { a.x = -2.0f * px; a.y = -2.0f * py; }
  else         { a.x = -2.0f * pz; a.y = 0.0f; }

  float best_val[8];
  int   best_idx[8];
#pragma unroll
  for (int v = 0; v < 8; ++v) { best_val[v] = 3.4e38f; best_idx[v] = 0; }

  for (int chunk = 0; chunk < PNPOS / 16; ++chunk) {
    const int j = chunk * 16 + m;   // candidate column this lane owns (N = lane&15)
    const float qx = positions[j * 3 + 0];
    const float qy = positions[j * 3 + 1];
    const float qz = positions[j * 3 + 2];
    // B 4x16 fp32 layout (mirror of A): lanes 0-15 hold K=0 (v0), K=1 (v1);
    // lanes 16-31 hold K=2 (v0), K=3 (v1); lane index = column N.
    v2f b;
    if (hi == 0) { b.x = qx; b.y = qy; }
    else         { b.x = qz; b.y = 0.0f; }
    const float pn = qx * qx + qy * qy + qz * qz;  // |p_j|^2 for this column

    v8f c = {};
    c = __builtin_amdgcn_wmma_f32_16x16x4_f32(
        /*neg_a=*/false, a, /*neg_b=*/false, b,
        /*c_mod=*/(short)0, c, /*reuse_a=*/false, /*reuse_b=*/false);

    // C/D layout: VGPR v, lanes 0-15 -> row M=v; lanes 16-31 -> row M=v+8.
    // Column N = lane&15 == m, so the score column index is exactly j.
#pragma unroll
    for (int v = 0; v < 8; ++v) {
      const float s = c[v] + pn;              // qn[m] constant per row: irrelevant to argmin
      if (s < best_val[v]) { best_val[v] = s; best_idx[v] = j; }
    }
  }

  // argmin across the 16 lanes of each half-wave (first-occurrence tie-break)
#pragma unroll
  for (int v = 0; v < 8; ++v) {
    float bv = best_val[v];
    int   bi = best_idx[v];
#pragma unroll
    for (int mask = 8; mask >= 1; mask >>= 1) {
      const float ov = __shfl_xor(bv, mask, 16);
      const int   oi = __shfl_xor(bi, mask, 16);
      if (ov < bv || (ov == bv && oi < bi)) { bv = ov; bi = oi; }
    }
    if (m == 0) idx_out[q0 + hi * 8 + v] = bi;
  }
}

// ---------------------------------------------------------------------------
// Kernel 2: out[b, c*T + t, :] = spatial[idx[b,c], :] + temporal[t, :]
// One block per (b,c). Spatial row (2 KB) cached in registers as float4;
// temporal table (128 KB) is L2-resident. Output (512 MB > 192 MB L2) is
// written with 128-bit non-temporal stores to stream past the caches.
// ---------------------------------------------------------------------------
__global__ __launch_bounds__(256) void bcast_add_pe(
    const float* __restrict__ spatial,    // (NPOS, E)
    const float* __restrict__ temporal,   // (T, E)
    const int*   __restrict__ idx,        // (B*C)
    float*       __restrict__ out)        // (B, C*T, E)
{
  const int bc  = blockIdx.x;             // 0 .. B*C-1
  const int id  = idx[bc];
  const int tid = threadIdx.x;
  const int e4  = tid & 127;              // float4 column within a row (E/4 = 128)
  const int t0  = tid >> 7;               // 0 or 1: two t-rows in flight per iter

  const v4f* __restrict__ srow = (const v4f*)(spatial + (size_t)id * PE);
  const v4f s = srow[e4];

  const v4f* __restrict__ trow = (const v4f*)temporal;
  v4f* __restrict__ orow = (v4f*)out + (size_t)bc * PT * (PE / 4);

#pragma unroll 4
  for (int t = t0; t < PT; t += 2) {
    const v4f tv = trow[t * (PE / 4) + e4];
    const v4f r  = s + tv;
    __builtin_nontemporal_store(r, &orow[t * (PE / 4) + e4]);
  }
}

// ---------------------------------------------------------------------------
extern "C" void kernel_launch(void* const* d_in, const int* in_sizes, int n_in,
                              void* d_out, int out_size, void* d_ws, size_t ws_size,
                              hipStream_t stream) {
  const float* pos        = (const float*)d_in[0];  // (B, C, 3)
  const float* positions  = (const float*)d_in[1];  // (NPOS, 3)
  const float* spatial    = (const float*)d_in[2];  // (NPOS, E)
  const float* temporal   = (const float*)d_in[3];  // (T, E)
  float*       out        = (float*)d_out;          // (B, C*T, E)
  int*         idx        = (int*)d_ws;             // B*C ints of scratch

  (void)in_sizes; (void)n_in; (void)out_size; (void)ws_size;

  nearest_idx_wmma<<<(PB * PC) / 16, 32, 0, stream>>>(pos, positions, idx);
  bcast_add_pe<<<PB * PC, 256, 0, stream>>>(spatial, temporal, idx, out);
}